// ISTFT_73667279061383
// MI455X (gfx1250) — compile-verified
//
#include <hip/hip_runtime.h>

typedef float v2f __attribute__((ext_vector_type(2)));
typedef float v8f __attribute__((ext_vector_type(8)));

#define NFFT   1024
#define KDIM   1026          // 513 freq bins * {re, im}
#define KP     1028          // padded to multiple of 4 for 16x16x4 WMMA
#define NB     64            // batch
#define NT     5             // only frames 0..4 contribute to samples [512,1112)
#define MROWS  (NB * NT)     // 320 GEMM rows
#define OUTLEN 600

// ---------------- workspace layout (in floats) ----------------
// [A_OFF, +320*1028)   gathered A matrix  (row = b*5+t, col = 2*f+c)
// [B_OFF, +1024*1028)  basis^T            (row = time n, col = 2*f+c)
// [F_OFF, +320*1024)   frames output of the GEMM
#define A_OFF  ((size_t)0)
#define B_OFF  ((size_t)MROWS * KP)
#define F_OFF  (B_OFF + (size_t)NFFT * KP)

// Build the inverse-rDFT basis, transposed: bT[n*KP + (2f+c)]
//   c==0 :  a_f * cos(2*pi*f*n/1024)
//   c==1 : -a_f * sin(2*pi*f*n/1024)   (0 for f==0, f==512: irfft ignores those imags)
//   a_f = 1/1024 for f in {0,512}, else 2/1024
__global__ void fill_basisT(float* __restrict__ bT) {
    int idx = blockIdx.x * blockDim.x + threadIdx.x;
    if (idx >= NFFT * KP) return;
    int n  = idx / KP;
    int kk = idx - n * KP;
    float v = 0.0f;
    if (kk < KDIM) {
        int f = kk >> 1;
        int c = kk & 1;
        float a = (f == 0 || f == 512) ? (1.0f / 1024.0f) : (2.0f / 1024.0f);
        int p = (f * n) & (NFFT - 1);                  // reduce angle mod 2*pi exactly
        float theta = (float)p * 6.13592315154256e-3f; // 2*pi/1024
        if (c == 0)  v = a * cosf(theta);
        else         v = (f == 0 || f == 512) ? 0.0f : -a * sinf(theta);
    }
    bT[idx] = v;
}

// Gather A[b*5+t][2f+c] = x[b, 0, f, t, c] from the (64,1,513,1024,2) input.
__global__ void gather_A(const float* __restrict__ x, float* __restrict__ Ag) {
    int idx = blockIdx.x * blockDim.x + threadIdx.x;
    if (idx >= MROWS * KP) return;
    int r  = idx / KP;
    int kk = idx - r * KP;
    float v = 0.0f;
    if (kk < KDIM) {
        int b = r / NT, t = r - b * NT;
        int f = kk >> 1, c = kk & 1;
        v = x[(((size_t)b * 513 + f) * 1024 + t) * 2 + c];
    }
    Ag[idx] = v;
}

// One wave (32 threads) computes one 16x16 f32 tile of frames = A (320xKP) * B (KPx1024)
// using V_WMMA_F32_16X16X4_F32.  ISA fragment layouts (7.12.2):
//   A: lane l -> row l%16, VGPR{0,1} -> K = 2*(l/16)+{0,1}    (B symmetric, col l%16)
//   C/D: lane l, VGPR v -> M = v + 8*(l/16), N = l%16
__global__ void __launch_bounds__(32) wmma_gemm(const float* __restrict__ Ag,
                                                const float* __restrict__ bT,
                                                float* __restrict__ frames) {
    const int ntile = blockIdx.x;      // 0..63  (N = 1024 / 16)
    const int mtile = blockIdx.y;      // 0..19  (M = 320  / 16)
    const int lane  = threadIdx.x;     // wave32
    const int half  = lane >> 4;
    const int l16   = lane & 15;

    const float* aRow = Ag + (size_t)(mtile * 16 + l16) * KP + 2 * half;
    const float* bRow = bT + (size_t)(ntile * 16 + l16) * KP + 2 * half;

    v8f acc = {};
    for (int k0 = 0; k0 < KP; k0 += 4) {
        v2f a = *(const v2f*)(aRow + k0);   // 8B aligned: offsets all even
        v2f b = *(const v2f*)(bRow + k0);
        acc = __builtin_amdgcn_wmma_f32_16x16x4_f32(
            /*neg_a=*/false, a, /*neg_b=*/false, b,
            /*c_mod=*/(short)0, acc, /*reuse_a=*/false, /*reuse_b=*/false);
    }

    float* out = frames + (size_t)(mtile * 16 + 8 * half) * NFFT + ntile * 16 + l16;
#pragma unroll
    for (int v = 0; v < 8; ++v)
        out[(size_t)v * NFFT] = acc[v];
}

// Overlap-add of the valid frames + window-envelope normalization.
// n = 512+m:  t=2 (k=m) always; t=1 (k=256+m<=855) always; t=0 iff m<=511;
//             t=3 iff m>=256; t=4 iff m>=512.  env = 3 for m<256 else 4.
__global__ void combine(const float* __restrict__ frames, float* __restrict__ y) {
    int idx = blockIdx.x * blockDim.x + threadIdx.x;
    if (idx >= NB * OUTLEN) return;
    int b = idx / OUTLEN;
    int m = idx - b * OUTLEN;
    const float* Fb = frames + (size_t)b * NT * NFFT;
    float s = Fb[2 * NFFT + m] + Fb[1 * NFFT + 256 + m];
    if (m <= 511) s += Fb[0 * NFFT + 512 + m];
    if (m >= 256) s += Fb[3 * NFFT + m - 256];
    if (m >= 512) s += Fb[4 * NFFT + m - 512];
    float inv = (m < 256) ? (1.0f / 3.0f) : 0.25f;
    y[idx] = s * inv;
}

extern "C" void kernel_launch(void* const* d_in, const int* in_sizes, int n_in,
                              void* d_out, int out_size, void* d_ws, size_t ws_size,
                              hipStream_t stream) {
    const float* x = (const float*)d_in[0];
    float* ws = (float*)d_ws;
    float* Ag     = ws + A_OFF;
    float* bT     = ws + B_OFF;
    float* frames = ws + F_OFF;
    float* y      = (float*)d_out;

    {   // basis: 1024*1028 entries
        int n = NFFT * KP;
        fill_basisT<<<(n + 255) / 256, 256, 0, stream>>>(bT);
    }
    {   // gather: 320*1028 entries (reads only ~1.3 MB of the 268 MB input)
        int n = MROWS * KP;
        gather_A<<<(n + 255) / 256, 256, 0, stream>>>(x, Ag);
    }
    {   // GEMM: one wave per 16x16 tile
        dim3 grid(NFFT / 16, MROWS / 16);   // 64 x 20
        wmma_gemm<<<grid, 32, 0, stream>>>(Ag, bT, frames);
    }
    {   // overlap-add + normalize -> (64, 1, 600)
        int n = NB * OUTLEN;
        combine<<<(n + 255) / 256, 256, 0, stream>>>(frames, y);
    }
}